// AdditiveAttention_58334245814472
// MI455X (gfx1250) — compile-verified
//
#include <hip/hip_runtime.h>
#include <hip/hip_bf16.h>

typedef __attribute__((ext_vector_type(2))) float v2f;
typedef __attribute__((ext_vector_type(8))) float v8f;

#define B_  4
#define LQ  512
#define LK  512
#define D_  256
#define H_  128

// ---------------------------------------------------------------------------
// fast tanh: gfx1250 has V_TANH_F32
// ---------------------------------------------------------------------------
__device__ __forceinline__ float fast_tanh(float x) {
#if __has_builtin(__builtin_amdgcn_tanhf)
    return __builtin_amdgcn_tanhf(x);
#else
    return tanhf(x);
#endif
}

// ---------------------------------------------------------------------------
// threefry2x32 with key (0,42)  (jax.random.key(42)), JAX split-half counters
// ---------------------------------------------------------------------------
__device__ __forceinline__ unsigned rotl32(unsigned x, int r) {
    return (x << r) | (x >> (32 - r));
}

__device__ __forceinline__ void threefry2x32_42(unsigned c0, unsigned c1,
                                                unsigned& o0, unsigned& o1) {
    const unsigned ks0 = 0u, ks1 = 42u, ks2 = 0x1BD11BDAu ^ 0u ^ 42u;
    unsigned x0 = c0 + ks0, x1 = c1 + ks1;
    const int R0[4] = {13, 15, 26, 6};
    const int R1[4] = {17, 29, 16, 24};
#pragma unroll
    for (int i = 0; i < 4; ++i) { x0 += x1; x1 = rotl32(x1, R0[i]); x1 ^= x0; }
    x0 += ks1; x1 += ks2 + 1u;
#pragma unroll
    for (int i = 0; i < 4; ++i) { x0 += x1; x1 = rotl32(x1, R1[i]); x1 ^= x0; }
    x0 += ks2; x1 += ks0 + 2u;
#pragma unroll
    for (int i = 0; i < 4; ++i) { x0 += x1; x1 = rotl32(x1, R0[i]); x1 ^= x0; }
    x0 += ks0; x1 += ks1 + 3u;
#pragma unroll
    for (int i = 0; i < 4; ++i) { x0 += x1; x1 = rotl32(x1, R1[i]); x1 ^= x0; }
    x0 += ks1; x1 += ks2 + 4u;
#pragma unroll
    for (int i = 0; i < 4; ++i) { x0 += x1; x1 = rotl32(x1, R0[i]); x1 ^= x0; }
    x0 += ks2; x1 += ks0 + 5u;
    o0 = x0; o1 = x1;
}

__device__ __forceinline__ float gumbel_at(unsigned j, unsigned half) {
    unsigned o0, o1, bits;
    if (j < half) { threefry2x32_42(j, j + half, o0, o1); bits = o0; }
    else          { threefry2x32_42(j - half, j, o0, o1); bits = o1; }
    float u = __uint_as_float((bits >> 9) | 0x3F800000u) - 1.0f;
    u = fmaxf(u, 1.1754944e-38f);
    return -__logf(-__logf(u));
}

// ---------------------------------------------------------------------------
// Kernel 1: projection GEMM via V_WMMA_F32_16X16X4_F32.
// Y = X[2048x256] * W[128x256]^T.  One 16x16 C-tile per wave, K-loop of 64.
// transposed==0 : Y row-major [2048,128]   (q projection)
// transposed==1 : Y as [B][H][Lk]          (k projection, for coalesced reads)
// ---------------------------------------------------------------------------
__global__ __launch_bounds__(256) void proj_wmma_kernel(
        const float* __restrict__ X, const float* __restrict__ W,
        float* __restrict__ Y, int transposed) {
    const int tid  = threadIdx.x;
    const int wave = tid >> 5;
    const int lane = tid & 31;
    const int tile = blockIdx.x * 8 + wave;   // 1024 tiles total
    const int tileM = tile >> 3;              // 128 row tiles
    const int tileN = tile & 7;               // 8 col tiles (H=128)
    const int hi = lane >> 4;                 // half-wave
    const int l  = lane & 15;

    // A frag: lane holds X[tileM*16+l][k0 + 2*hi + {0,1}]
    const float* aptr = X + (size_t)(tileM * 16 + l) * D_ + 2 * hi;
    // B frag: lane holds W[tileN*16+l][k0 + 2*hi + {0,1}]  (B[k][n] = W[n][k])
    const float* bptr = W + (size_t)(tileN * 16 + l) * D_ + 2 * hi;

    v8f acc = {};
#pragma unroll 8
    for (int k0 = 0; k0 < D_; k0 += 4) {
        v2f a = *(const v2f*)(aptr + k0);
        v2f b = *(const v2f*)(bptr + k0);
        acc = __builtin_amdgcn_wmma_f32_16x16x4_f32(
                false, a, false, b, (short)0, acc, false, false);
    }

    // D layout: vgpr v, lane -> C[M = v + 8*hi][N = l]
    if (!transposed) {
        float* dst = Y + (size_t)(tileM * 16 + hi * 8) * H_ + tileN * 16 + l;
#pragma unroll
        for (int v = 0; v < 8; ++v) dst[(size_t)v * H_] = acc[v];
    } else {
        const int n = tileN * 16 + l;
#pragma unroll
        for (int v = 0; v < 8; ++v) {
            int gm = tileM * 16 + v + hi * 8;      // global key row
            int b  = gm >> 9;                      // /512
            int kk = gm & 511;
            Y[(size_t)b * (H_ * LK) + (size_t)n * LK + kk] = acc[v];
        }
    }
}

// ---------------------------------------------------------------------------
// Kernel 2: per (b,q) block of 512 threads (one per k).
// acc = sum_h tanh(qp[h] + kpT[b][h][k]) * (Wv0[h]-Wv1[h])
// out = mask<2 ? 1 : sigmoid((acc + g0 - g1)/tau)
// ---------------------------------------------------------------------------
__global__ __launch_bounds__(512) void attn_scores_kernel(
        const float* __restrict__ qp, const float* __restrict__ kpT,
        const int* __restrict__ mask, const float* __restrict__ Wv,
        float* __restrict__ out) {
    const int bq = blockIdx.x;          // b*512 + q
    const int b  = bq >> 9;
    const int tid = threadIdx.x;        // = k
    __shared__ float s_q[H_];
    __shared__ float s_wvd[H_];
    if (tid < H_) {
        s_q[tid]   = qp[(size_t)bq * H_ + tid];
        s_wvd[tid] = Wv[tid] - Wv[H_ + tid];
    }
    __syncthreads();

    const float* kcol = kpT + (size_t)b * (H_ * LK) + tid;
    float acc = 0.0f;
#pragma unroll 8
    for (int h = 0; h < H_; ++h) {
        float t = fast_tanh(s_q[h] + kcol[(size_t)h * LK]);
        acc = __builtin_fmaf(t, s_wvd[h], acc);
    }

    const unsigned base = (unsigned)bq * (unsigned)LK + (unsigned)tid;
    const unsigned half = (unsigned)(B_ * LQ * LK);   // n/2 of [B,Lq,Lk,2]
    float g0 = gumbel_at(2u * base,      half);
    float g1 = gumbel_at(2u * base + 1u, half);

    float d = (acc + g0 - g1) * 100.0f;               // /tau, tau = 0.01
    float w = 1.0f / (1.0f + __expf(-d));
    out[base] = (mask[base] < 2) ? 1.0f : w;
}

// ---------------------------------------------------------------------------
extern "C" void kernel_launch(void* const* d_in, const int* in_sizes, int n_in,
                              void* d_out, int out_size, void* d_ws, size_t ws_size,
                              hipStream_t stream) {
    const float* queries = (const float*)d_in[0];
    const float* keys    = (const float*)d_in[1];
    const int*   mask    = (const int*)  d_in[2];
    const float* Wq      = (const float*)d_in[3];
    const float* Wk      = (const float*)d_in[4];
    const float* Wv      = (const float*)d_in[5];
    float* out = (float*)d_out;

    float* qp  = (float*)d_ws;                 // [B*Lq, H]   = 1 MB
    float* kpT = qp + (size_t)B_ * LQ * H_;    // [B, H, Lk]  = 1 MB

    // 1024 tiles / 8 waves per block -> 128 blocks
    proj_wmma_kernel<<<128, 256, 0, stream>>>(queries, Wq, qp, 0);
    proj_wmma_kernel<<<128, 256, 0, stream>>>(keys,    Wk, kpT, 1);
    attn_scores_kernel<<<B_ * LQ, 512, 0, stream>>>(qp, kpT, mask, Wv, out);
}